// MetricalGNN_82308753260711
// MI455X (gfx1250) — compile-verified
//
#include <hip/hip_runtime.h>
#include <hip/hip_bf16.h>
#include <math.h>

// ---------------- problem constants (match reference) ----------------
#define N_NODES 50000
#define N_EDGES 600000
#define F_DIM   128
#define N_TYPES 4
#define NSTRIPS (N_NODES / 16)        // 3125 strips of 16 rows
#define STRIPS_PER_BLOCK 4
#define KCHUNK 64                     // stage 64 rows of W (32KB) in LDS at a time

typedef float v2f __attribute__((ext_vector_type(2)));
typedef float v8f __attribute__((ext_vector_type(8)));
typedef float f4v __attribute__((ext_vector_type(4)));

// ---------------------------------------------------------------------
// Combine per-type weights:
//   wcomb[0]    = (1/4) * sum_t W_self[t]            [128x128]
//   wcomb[1+t]  = (1/4) * W_neigh[t]                 [128x128] x4
//   bcomb       = (1/4) * sum_t b[t]                 [128]
// ---------------------------------------------------------------------
__global__ void prep_weights(const float* __restrict__ Wself,
                             const float* __restrict__ Wneigh,
                             const float* __restrict__ b,
                             float* __restrict__ wcomb,
                             float* __restrict__ bcomb) {
    int idx = blockIdx.x * blockDim.x + threadIdx.x;
    if (idx < F_DIM * F_DIM) {
        float s = 0.f;
        #pragma unroll
        for (int t = 0; t < N_TYPES; ++t) s += Wself[t * F_DIM * F_DIM + idx];
        wcomb[idx] = s * 0.25f;
        #pragma unroll
        for (int t = 0; t < N_TYPES; ++t)
            wcomb[(t + 1) * F_DIM * F_DIM + idx] = Wneigh[t * F_DIM * F_DIM + idx] * 0.25f;
    }
    if (idx < F_DIM) {
        float s = 0.f;
        #pragma unroll
        for (int t = 0; t < N_TYPES; ++t) s += b[t * F_DIM + idx];
        bcomb[idx] = s * 0.25f;
    }
}

__global__ void zero_buf(float* __restrict__ p, int n) {
    int i = blockIdx.x * blockDim.x + threadIdx.x;
    if (i < n) p[i] = 0.f;
}

// ---------------------------------------------------------------------
// Scatter pass for one edge type: one wave per edge.
// acc[dst] += h[src] (512B coalesced gather + f32 global atomics), cnt[dst] += 1
// ---------------------------------------------------------------------
__global__ void scatter_type(const float* __restrict__ h,
                             const int* __restrict__ edge_index,
                             const int* __restrict__ edge_type,
                             float* __restrict__ acc,
                             float* __restrict__ cnt,
                             int t) {
    int wave = (blockIdx.x * blockDim.x + threadIdx.x) >> 5;
    int lane = threadIdx.x & 31;
    if (wave >= N_EDGES) return;
    if (edge_type[wave] != t) return;           // wave-uniform
    int src = edge_index[wave];                 // row 0 of edge_index
    int dst = edge_index[N_EDGES + wave];       // row 1 of edge_index
    const f4v* hs = (const f4v*)(h + (size_t)src * F_DIM);
    f4v v = hs[lane];
    float* ad = acc + (size_t)dst * F_DIM + lane * 4;
    atomicAdd(&ad[0], v.x);
    atomicAdd(&ad[1], v.y);
    atomicAdd(&ad[2], v.z);
    atomicAdd(&ad[3], v.w);
    if (lane == 0) atomicAdd(&cnt[dst], 1.0f);
}

// ---------------------------------------------------------------------
// WMMA f32 GEMM:  C[strip rows x 128] (+)= (X * rowscale) @ W  [+ bias]
//   - one wave computes a 16-row x 128-col strip (8 x v8f accumulators)
//   - W staged through LDS in 32KB chunks (cooperative across 4 waves)
//   - rowscale = 1/max(cnt,1) folded into the A-fragment load (mean)
// Fragment layouts per CDNA5 ISA 7.12.2:
//   A (16x4 f32): VGPR v, lanes 0-15 -> k=v, lanes 16-31 -> k=v+2 (row = lane&15)
//   B (4x16 f32): VGPR v, lanes 0-15 -> k=v (n=lane), lanes 16-31 -> k=v+2
//   C (16x16 f32): VGPR r, lanes 0-15 -> row r, lanes 16-31 -> row r+8
// ---------------------------------------------------------------------
__global__ void gemm_wmma(const float* __restrict__ X,
                          const float* __restrict__ Wmat,
                          const float* __restrict__ cnt,   // nullptr -> scale 1
                          const float* __restrict__ bias,  // nullptr -> 0
                          float* __restrict__ C,
                          int accumulate) {
    __shared__ float Ws[KCHUNK * F_DIM];   // 32 KB

    int tid  = threadIdx.x;                // 0..127 (4 waves)
    int wave = tid >> 5;
    int lane = tid & 31;
    int strip = blockIdx.x * STRIPS_PER_BLOCK + wave;
    bool active = (strip < NSTRIPS);       // wave-uniform guard (EXEC stays all-1s)

    int row0 = strip * 16;
    int r    = lane & 15;                  // A row within strip / B-C column index
    int ko   = (lane >> 4) * 2;            // k sub-offset for A/B fragments
    int rhalf = (lane >> 4) * 8;           // C row-half offset

    float inv = 1.0f;
    if (active && cnt) inv = 1.0f / fmaxf(cnt[row0 + r], 1.0f);

    v8f acc[8];
    if (active) {
        if (accumulate) {
            #pragma unroll
            for (int ct = 0; ct < 8; ++ct)
                #pragma unroll
                for (int rr = 0; rr < 8; ++rr)
                    acc[ct][rr] = C[(size_t)(row0 + rhalf + rr) * F_DIM + ct * 16 + r];
        } else {
            #pragma unroll
            for (int ct = 0; ct < 8; ++ct) {
                float bv = bias ? bias[ct * 16 + r] : 0.f;
                #pragma unroll
                for (int rr = 0; rr < 8; ++rr) acc[ct][rr] = bv;
            }
        }
    }

    const float* xrow = X + (size_t)(row0 + r) * F_DIM;

    for (int kc = 0; kc < F_DIM; kc += KCHUNK) {
        // cooperative LDS stage of W rows [kc, kc+KCHUNK)
        const f4v* wg = (const f4v*)(Wmat + (size_t)kc * F_DIM);
        f4v* wl = (f4v*)Ws;
        for (int i = tid; i < KCHUNK * F_DIM / 4; i += 128) wl[i] = wg[i];
        __syncthreads();

        if (active) {
            for (int k0 = 0; k0 < KCHUNK; k0 += 4) {
                v2f a;
                a.x = xrow[kc + k0 + ko]     * inv;
                a.y = xrow[kc + k0 + ko + 1] * inv;
                #pragma unroll
                for (int ct = 0; ct < 8; ++ct) {
                    v2f b;
                    b.x = Ws[(k0 + ko)     * F_DIM + ct * 16 + r];
                    b.y = Ws[(k0 + ko + 1) * F_DIM + ct * 16 + r];
                    acc[ct] = __builtin_amdgcn_wmma_f32_16x16x4_f32(
                        false, a, false, b, (short)0, acc[ct], false, false);
                }
            }
        }
        __syncthreads();
    }

    if (active) {
        #pragma unroll
        for (int ct = 0; ct < 8; ++ct)
            #pragma unroll
            for (int rr = 0; rr < 8; ++rr)
                C[(size_t)(row0 + rhalf + rr) * F_DIM + ct * 16 + r] = acc[ct][rr];
    }
}

// ---------------------------------------------------------------------
// Row-wise L2 normalize (clamped at 1e-12) + ReLU. One wave per node.
// ---------------------------------------------------------------------
__global__ void norm_relu(const float* __restrict__ in, float* __restrict__ out) {
    int wave = (blockIdx.x * blockDim.x + threadIdx.x) >> 5;
    int lane = threadIdx.x & 31;
    if (wave >= N_NODES) return;
    const f4v* ip = (const f4v*)(in + (size_t)wave * F_DIM);
    f4v v = ip[lane];
    float s = v.x * v.x + v.y * v.y + v.z * v.z + v.w * v.w;
    #pragma unroll
    for (int off = 16; off > 0; off >>= 1) s += __shfl_xor(s, off, 32);
    float scale = 1.0f / fmaxf(sqrtf(s), 1e-12f);
    v.x = fmaxf(v.x * scale, 0.f);
    v.y = fmaxf(v.y * scale, 0.f);
    v.z = fmaxf(v.z * scale, 0.f);
    v.w = fmaxf(v.w * scale, 0.f);
    ((f4v*)(out + (size_t)wave * F_DIM))[lane] = v;
}

// ---------------------------------------------------------------------
extern "C" void kernel_launch(void* const* d_in, const int* in_sizes, int n_in,
                              void* d_out, int out_size, void* d_ws, size_t ws_size,
                              hipStream_t stream) {
    const float* x       = (const float*)d_in[0];
    const float* Wself1  = (const float*)d_in[1];
    const float* Wneigh1 = (const float*)d_in[2];
    const float* b1      = (const float*)d_in[3];
    const float* Wself2  = (const float*)d_in[4];
    const float* Wneigh2 = (const float*)d_in[5];
    const float* b2      = (const float*)d_in[6];
    const int*   eidx    = (const int*)d_in[7];
    const int*   etype   = (const int*)d_in[8];
    float* out = (float*)d_out;

    // workspace layout (floats)
    float* ws    = (float*)d_ws;
    float* acc   = ws;                                    // N*F
    float* cnt   = acc + (size_t)N_NODES * F_DIM;         // N
    float* h1    = cnt + N_NODES;                         // N*F
    float* wcomb = h1 + (size_t)N_NODES * F_DIM;          // 5*F*F
    float* bcomb = wcomb + 5 * F_DIM * F_DIM;             // F

    const int ZN = N_NODES * F_DIM + N_NODES;
    dim3 zgrid((ZN + 255) / 256);
    dim3 ggrid((NSTRIPS + STRIPS_PER_BLOCK - 1) / STRIPS_PER_BLOCK);   // 782
    dim3 sgrid((N_EDGES * 32 + 255) / 256);                            // 75000
    dim3 ngrid((N_NODES * 32 + 255) / 256);                            // 6250

    // ---------------- layer 1: raw output accumulates in d_out ----------------
    prep_weights<<<64, 256, 0, stream>>>(Wself1, Wneigh1, b1, wcomb, bcomb);
    gemm_wmma<<<ggrid, 128, 0, stream>>>(x, wcomb, nullptr, bcomb, out, 0);
    for (int t = 0; t < N_TYPES; ++t) {
        zero_buf<<<zgrid, 256, 0, stream>>>(acc, ZN);
        scatter_type<<<sgrid, 256, 0, stream>>>(x, eidx, etype, acc, cnt, t);
        gemm_wmma<<<ggrid, 128, 0, stream>>>(acc, wcomb + (t + 1) * F_DIM * F_DIM,
                                             cnt, nullptr, out, 1);
    }
    norm_relu<<<ngrid, 256, 0, stream>>>(out, h1);

    // ---------------- layer 2: final output in d_out ----------------
    prep_weights<<<64, 256, 0, stream>>>(Wself2, Wneigh2, b2, wcomb, bcomb);
    gemm_wmma<<<ggrid, 128, 0, stream>>>(h1, wcomb, nullptr, bcomb, out, 0);
    for (int t = 0; t < N_TYPES; ++t) {
        zero_buf<<<zgrid, 256, 0, stream>>>(acc, ZN);
        scatter_type<<<sgrid, 256, 0, stream>>>(h1, eidx, etype, acc, cnt, t);
        gemm_wmma<<<ggrid, 128, 0, stream>>>(acc, wcomb + (t + 1) * F_DIM * F_DIM,
                                             cnt, nullptr, out, 1);
    }
}